// ClassLoss_22728966930467
// MI455X (gfx1250) — compile-verified
//
#include <hip/hip_runtime.h>

typedef __attribute__((ext_vector_type(2))) float v2f;
typedef __attribute__((ext_vector_type(8))) float v8f;

#define NUM_CLS 8
#define MAX_INST 64
#define BATCH 32
#define KLAB 64
#define HW (1024 * 1024)
#define BPB 64                 // histogram blocks per batch sample
#define CHUNK (HW / BPB)       // 16384 pixels per block

// ---------------- zero the per-batch histograms in workspace ----------------
__global__ void zero_hist(unsigned int* __restrict__ hist) {
    for (int i = threadIdx.x; i < BATCH * MAX_INST; i += blockDim.x) hist[i] = 0u;
}

// ---------------- bandwidth kernel: 128MB stream -> per-batch 64-bin hist ---
__global__ __launch_bounds__(256) void hist_kernel(const int* __restrict__ mask,
                                                   unsigned int* __restrict__ hist) {
    __shared__ unsigned int h[8][MAX_INST];   // per-wave privatized histograms
    const int wave = threadIdx.x >> 5;
    for (int i = threadIdx.x; i < 8 * MAX_INST; i += 256) ((unsigned int*)h)[i] = 0u;
    __syncthreads();

    const int b     = blockIdx.x / BPB;
    const int chunk = blockIdx.x % BPB;
    const int4* m4  = (const int4*)(mask + (long long)b * HW + (long long)chunk * CHUNK);

    for (int i = threadIdx.x; i < CHUNK / 4; i += 256) {
        int4 v = m4[i];                        // global_load_b128
        atomicAdd(&h[wave][v.x & 63], 1u);     // ds_add_u32 (64 bins == 64 banks)
        atomicAdd(&h[wave][v.y & 63], 1u);
        atomicAdd(&h[wave][v.z & 63], 1u);
        atomicAdd(&h[wave][v.w & 63], 1u);
    }
    __syncthreads();

    if (threadIdx.x < MAX_INST) {
        unsigned int s = 0;
        #pragma unroll
        for (int w = 0; w < 8; ++w) s += h[w][threadIdx.x];
        atomicAdd(&hist[b * MAX_INST + threadIdx.x], s);   // integer -> deterministic
    }
}

// ---------------- finish: WMMA einsum + BCEWithLogits reduction -------------
// Per batch: cnts = O^T (8x64, zero-padded to 16x64) x sizes (64), computed as
// 16 chained V_WMMA_F32_16X16X4_F32 with B = sizes chunk broadcast to all cols.
__global__ __launch_bounds__(256) void finish_kernel(const float* __restrict__ pred,
                                                     const int* __restrict__ lab,  // int32 view of int64 label_gt
                                                     const unsigned int* __restrict__ hist,
                                                     float* __restrict__ out) {
    __shared__ float partial[8];
    const int wave = threadIdx.x >> 5;
    const int lane = threadIdx.x & 31;
    const int row  = lane & 15;          // A-matrix M index held by this lane
    const int hi   = (lane >> 4) & 1;    // lanes 16-31 hold K=2,3

    float acc = 0.0f;
    for (int it = 0; it < 4; ++it) {
        const int b = wave + 8 * it;     // 8 waves x 4 iterations = 32 batches
        v8f c = {0.f, 0.f, 0.f, 0.f, 0.f, 0.f, 0.f, 0.f};
        #pragma unroll
        for (int t = 0; t < 16; ++t) {
            const int j0 = 4 * t + (hi ? 2 : 0);
            // label_gt[b][j][s] is int64; low dword of (b,j,s) at index ((b*64+j)*2+s)*2
            const int base = (b * KLAB + j0) * 4;
            const int id0 = lab[base + 0] & 63;
            const int lb0 = lab[base + 2];
            const int id1 = lab[base + 4] & 63;
            const int lb1 = lab[base + 6];
            v2f a, bb;
            // A[row, k] = valid(label)*onehot: 1 iff label == row+1 and row < 8
            a.x = (lb0 == row + 1 && row < NUM_CLS) ? 1.0f : 0.0f;
            a.y = (lb1 == row + 1 && row < NUM_CLS) ? 1.0f : 0.0f;
            // B[k, n] = sizes[j] for all n (broadcast column vector)
            bb.x = (float)hist[b * MAX_INST + id0];
            bb.y = (float)hist[b * MAX_INST + id1];
            c = __builtin_amdgcn_wmma_f32_16x16x4_f32(
                    false, a, false, bb, (short)0, c, false, false);
        }
        // C/D layout: lanes 0-15, VGPR v holds D[M=v, N=lane]; all N cols equal,
        // so lane 0 sees cnts[c] in c[c] for c = 0..7.
        if (lane == 0) {
            float s = 0.0f;
            #pragma unroll
            for (int cc = 0; cc < NUM_CLS; ++cc) {
                float y = fminf(c[cc] * (1.0f / 100.0f), 1.0f);
                float x = pred[b * NUM_CLS + cc];
                s += fmaxf(x, 0.0f) - x * y + log1pf(expf(-fabsf(x)));
            }
            acc += s;
        }
    }
    if (lane == 0) partial[wave] = acc;
    __syncthreads();
    if (threadIdx.x == 0) {
        float s = 0.0f;
        #pragma unroll
        for (int w = 0; w < 8; ++w) s += partial[w];
        out[0] = s * (1.0f / (BATCH * NUM_CLS));
    }
}

extern "C" void kernel_launch(void* const* d_in, const int* in_sizes, int n_in,
                              void* d_out, int out_size, void* d_ws, size_t ws_size,
                              hipStream_t stream) {
    const float* pred = (const float*)d_in[0];       // [32, 8] f32
    const int*   mask = (const int*)d_in[1];         // [32, 1024, 1024] i32
    const int*   lab  = (const int*)d_in[2];         // [32, 64, 2] i64 viewed as i32 pairs
    float* out = (float*)d_out;
    unsigned int* hist = (unsigned int*)d_ws;        // 32*64 u32 = 8KB

    zero_hist<<<1, 256, 0, stream>>>(hist);
    hist_kernel<<<BATCH * BPB, 256, 0, stream>>>(mask, hist);
    finish_kernel<<<1, 256, 0, stream>>>(pred, lab, hist, out);
}